// Generator_6433861009464
// MI455X (gfx1250) — compile-verified
//
#include <hip/hip_runtime.h>
#include <hip/hip_bf16.h>
#include <math.h>

#define B_   32
#define L_   1024
#define D_   1024
#define E_   100
#define EP   101          // E + 1
#define F_   1025         // D + 1
#define PSTRIDE 1040      // padded row stride for ws_proj (65 tiles of 16)
#define BEP  (B_ * EP)    // 3232 rows

typedef float v2f __attribute__((ext_vector_type(2)));
typedef float v8f __attribute__((ext_vector_type(8)));

// D = A(16x4, f32) * B(4x16, f32) + C(16x16, f32)
__device__ __forceinline__ v8f wmma_f32(v2f a, v2f b, v8f c) {
    return __builtin_amdgcn_wmma_f32_16x16x4_f32(
        /*neg_a=*/false, a, /*neg_b=*/false, b,
        /*c_mod=*/(short)0, c, /*reuse_a=*/false, /*reuse_b=*/false);
}

// ---------------- K1: ragged entity segment-sum -> ws_enc [B*EP, D] ----------
__global__ void entity_sum_kernel(const float* __restrict__ q_enc,
                                  const float* __restrict__ none_entity,
                                  const int* __restrict__ entity_ranges,
                                  float* __restrict__ ws_enc) {
    int row = blockIdx.x;            // 0 .. B*EP-1
    int b = row / EP, e = row % EP;
    int t = threadIdx.x;             // 256 threads, 4 columns each
    float acc0 = 0.f, acc1 = 0.f, acc2 = 0.f, acc3 = 0.f;
    if (e == E_) {
        acc0 = none_entity[t];
        acc1 = none_entity[t + 256];
        acc2 = none_entity[t + 512];
        acc3 = none_entity[t + 768];
    } else {
        int r0 = entity_ranges[((size_t)b * E_ + e) * 2 + 0];
        int r1 = entity_ranges[((size_t)b * E_ + e) * 2 + 1];
        for (int tok = r0 + 1; tok < r1; ++tok) {
            const float* src = q_enc + ((size_t)b * L_ + tok) * D_;
            acc0 += src[t];
            acc1 += src[t + 256];
            acc2 += src[t + 512];
            acc3 += src[t + 768];
        }
    }
    float* dst = ws_enc + (size_t)row * D_;
    dst[t]       = acc0;
    dst[t + 256] = acc1;
    dst[t + 512] = acc2;
    dst[t + 768] = acc3;
}

// --------- K2: ent_proj = ws_enc @ W^T + b  (M=3232, K=1024, N=1025) ---------
// Each wave computes a 16(M) x 64(N) tile: 1 A-fragment reused over 4 N-tiles.
// Out-of-range columns load a clamped (valid) W row unconditionally; their
// accumulator columns are simply not stored in the epilogue.
__global__ void proj_gemm_kernel(const float* __restrict__ enc,
                                 const float* __restrict__ W,     // [F_, D_] row-major
                                 const float* __restrict__ bias,  // [F_]
                                 float* __restrict__ proj) {      // [BEP, PSTRIDE]
    const int NT = PSTRIDE / 16;   // 65 N-tiles
    const int NG = 17;             // groups of 4 N-tiles (ceil(65/4))
    const int MT = BEP / 16;       // 202 M-tiles
    int wave = blockIdx.x * (blockDim.x >> 5) + (threadIdx.x >> 5);
    if (wave >= MT * NG) return;   // uniform per wave -> EXEC all 1s below
    int mt = wave / NG;
    int ng = wave % NG;
    int lane = threadIdx.x & 31;
    int lm = lane & 15;            // M (A), N (B/C) lane index
    int lh = lane >> 4;            // half-wave selector

    const float* aptr = enc + ((size_t)(mt * 16 + lm)) * D_;

    const float* bp[4];
    for (int j = 0; j < 4; ++j) {
        int n = ng * 64 + j * 16 + lm;       // column f == row of W
        int nc = n < F_ ? n : (F_ - 1);      // clamp: OOR lanes -> discarded cols
        bp[j] = W + (size_t)nc * D_;
    }

    v8f acc[4] = {};
    for (int k0 = 0; k0 < D_; k0 += 4) {
        v2f a;                               // A[m=lm][k = v + 2*lh]
        a.x = aptr[k0 + 2 * lh];
        a.y = aptr[k0 + 1 + 2 * lh];
        for (int j = 0; j < 4; ++j) {
            v2f bf;                          // B[k = v + 2*lh][n = lm] = W[n][k]
            bf.x = bp[j][k0 + 2 * lh];
            bf.y = bp[j][k0 + 1 + 2 * lh];
            acc[j] = wmma_f32(a, bf, acc[j]);
        }
    }
    for (int j = 0; j < 4; ++j) {
        int nt = ng * 4 + j;
        if (nt >= NT) break;
        int f = nt * 16 + lm;                // C/D: n = lm
        if (f >= F_) continue;
        float bb = bias[f];
        for (int v = 0; v < 8; ++v) {
            int m = v + 8 * lh;              // C/D: m = v + 8*lh
            proj[((size_t)(mt * 16 + m)) * PSTRIDE + f] = acc[j][v] + bb;
        }
    }
}

// -------- K3: fused scores GEMM + softmax + masked column accumulation -------
// grid = (13, B): 16 question rows per block, covers l < 208 >= Q_MAX.
// block = 128 (4 waves); wave w handles entity tiles {w, w+4} (7 tiles total).
__global__ void score_softmax_kernel(const float* __restrict__ q_enc,
                                     const float* __restrict__ hint,     // [B, L]
                                     const float* __restrict__ proj,     // [BEP, PSTRIDE]
                                     const int* __restrict__ q_q_len,    // [B]
                                     float* __restrict__ co_att,         // [B, L, EP]
                                     float* __restrict__ ws_att) {       // [B, EP]
    __shared__ float smem[16][EP + 3];
    __shared__ float rowsum[16];
    __shared__ float colsum[EP];

    int b  = blockIdx.y;
    int l0 = blockIdx.x * 16;
    int t  = threadIdx.x;
    int wave = t >> 5, lane = t & 31, lm = lane & 15, lh = lane >> 4;
    if (t < EP) colsum[t] = 0.f;

    const float* aptr = q_enc + ((size_t)b * L_ + (l0 + lm)) * D_;

    v8f acc[2] = {};
    const float* bp[2];
    int e0arr[2];
    int ntiles = 0;
    for (int j = 0; j < 2; ++j) {
        int nt = wave + 4 * j;
        if (nt < 7) {                        // uniform within a wave
            int e = nt * 16 + lm;
            int ec = e < EP ? e : E_;        // clamp: OOR lanes -> discarded cols
            e0arr[ntiles] = nt * 16;
            bp[ntiles] = proj + ((size_t)(b * EP + ec)) * PSTRIDE;
            ++ntiles;
        }
    }

    for (int k0 = 0; k0 < D_; k0 += 4) {     // f = 0..1023; f=1024 in epilogue
        v2f a;
        a.x = aptr[k0 + 2 * lh];
        a.y = aptr[k0 + 1 + 2 * lh];
        for (int j = 0; j < ntiles; ++j) {
            v2f bf;
            bf.x = bp[j][k0 + 2 * lh];
            bf.y = bp[j][k0 + 1 + 2 * lh];
            acc[j] = wmma_f32(a, bf, acc[j]);
        }
    }

    // epilogue: rank-1 hint term  scores += hint[l] * proj[e, 1024]
    float hv[8];
    for (int v = 0; v < 8; ++v)
        hv[v] = hint[(size_t)b * L_ + (l0 + v + 8 * lh)];
    for (int j = 0; j < ntiles; ++j) {
        int e = e0arr[j] + lm;
        if (e < EP) {
            float pf = bp[j][D_];            // proj[e, 1024] (bp already clamped==e)
            for (int v = 0; v < 8; ++v) {
                int m = v + 8 * lh;
                smem[m][e] = acc[j][v] + hv[v] * pf;
            }
        }
    }
    __syncthreads();

    if (t < 16) {                            // per-row softmax over 101 entities
        float mx = -3.402823466e38f;
        for (int e = 0; e < EP; ++e) mx = fmaxf(mx, smem[t][e]);
        float s = 0.f;
        for (int e = 0; e < EP; ++e) {
            float v = __expf(smem[t][e] - mx);
            smem[t][e] = v;
            s += v;
        }
        rowsum[t] = s;
    }
    __syncthreads();

    int qlen = q_q_len[b];
    for (int idx = t; idx < 16 * EP; idx += 128) {
        int row = idx / EP, e = idx % EP;
        int l = l0 + row;
        if (l < qlen) {                      // masked rows stay at memset 0
            float v = smem[row][e] / rowsum[row];
            co_att[((size_t)b * L_ + l) * EP + e] = v;
            atomicAdd(&colsum[e], v);        // LDS ds_add_f32
        }
    }
    __syncthreads();
    if (t < EP) atomicAdd(&ws_att[b * EP + t], colsum[t]);
}

// ---------------- K4: att = min(colsum, 1), drop none-entity column ----------
__global__ void finalize_att_kernel(const float* __restrict__ ws_att,
                                    float* __restrict__ att) {
    int i = blockIdx.x * blockDim.x + threadIdx.x;
    if (i >= B_ * E_) return;
    int b = i / E_, e = i % E_;
    att[i] = fminf(ws_att[b * EP + e], 1.0f);
}

extern "C" void kernel_launch(void* const* d_in, const int* in_sizes, int n_in,
                              void* d_out, int out_size, void* d_ws, size_t ws_size,
                              hipStream_t stream) {
    (void)in_sizes; (void)n_in; (void)ws_size;
    const float* q_enc         = (const float*)d_in[0];
    const float* hint          = (const float*)d_in[1];
    const float* W_attn        = (const float*)d_in[2];
    const float* b_attn        = (const float*)d_in[3];
    const float* none_entity   = (const float*)d_in[4];
    const int*   q_q_len       = (const int*)d_in[5];
    const int*   entity_ranges = (const int*)d_in[6];

    float* att    = (float*)d_out;               // [B, E]   = 3200
    float* co_att = att + (size_t)B_ * E_;       // [B,L,EP] = 3,309,568

    float* ws_enc  = (float*)d_ws;                        // [BEP, D_]
    float* ws_proj = ws_enc + (size_t)BEP * D_;           // [BEP, PSTRIDE]
    float* ws_att  = ws_proj + (size_t)BEP * PSTRIDE;     // [B, EP] col sums

    // zero outputs (masked co_att rows rely on this) and column accumulators
    hipMemsetAsync(d_out, 0, (size_t)out_size * sizeof(float), stream);
    hipMemsetAsync(ws_att, 0, (size_t)BEP * sizeof(float), stream);

    entity_sum_kernel<<<BEP, 256, 0, stream>>>(q_enc, none_entity,
                                               entity_ranges, ws_enc);

    {   // 202 M-tiles * 17 groups = 3434 waves, 4 waves per block
        int waves = (BEP / 16) * 17;
        int blocks = (waves + 3) / 4;
        proj_gemm_kernel<<<blocks, 128, 0, stream>>>(ws_enc, W_attn, b_attn,
                                                     ws_proj);
    }

    {   // 13 row-blocks of 16 cover l < 208 >= Q_MAX=200
        dim3 grid(13, B_);
        score_softmax_kernel<<<grid, 128, 0, stream>>>(q_enc, hint, ws_proj,
                                                       q_q_len, co_att, ws_att);
    }

    finalize_att_kernel<<<(B_ * E_ + 255) / 256, 256, 0, stream>>>(ws_att, att);
}